// BertEmbedding_1623497638029
// MI455X (gfx1250) — compile-verified
//
#include <hip/hip_runtime.h>
#include <cstdint>

// Problem constants (from reference): BZ=64, P=512, T=256, H=768
#define BZv     64
#define Pv      512
#define Tv      256
#define Hv      768
#define H4v     (Hv / 4)       // 192 float4 per hidden row
#define TILE    8              // tokens per gather block
#define MAXROWS (2 * TILE)     // max staged subword rows per tile
#define BLOCK   256

typedef unsigned int u32x4 __attribute__((ext_vector_type(4)));
typedef int          i32x4 __attribute__((ext_vector_type(4)));
typedef int          i32x8 __attribute__((ext_vector_type(8)));

// ---------------------------------------------------------------------------
// TDM gather kernel (primary path).
// One TENSOR_LOAD_TO_LDS in gather mode per 8-token tile: DMA engine gathers
// the <=16 subword rows (16-bit sorted indices) of enc_out[b] compacted into
// LDS; waves wait on TENSORcnt, then average 1-2 rows per token and store.
// ---------------------------------------------------------------------------
__global__ __launch_bounds__(BLOCK) void bert_gather_tdm_kernel(
        const float* __restrict__ enc,
        const int*   __restrict__ lens,
        const int*   __restrict__ start,
        float*       __restrict__ out) {
    __shared__ float4 rows_lds[MAXROWS][H4v];   // 16 * 3072B = 48 KB
    __shared__ int s_start[TILE];
    __shared__ int s_len[TILE];
    __shared__ int s_off[TILE];                 // intra-tile compacted row offset

    const int b   = blockIdx.y;
    const int t0  = blockIdx.x * TILE;
    const int tid = threadIdx.x;

    if (tid < TILE) {
        s_start[tid] = start[b * Tv + t0 + tid];
        s_len[tid]   = lens[b * Tv + t0 + tid];
    }
    __syncthreads();

    if (tid == 0) {                  // compacted-row offsets for the reducer
        int o = 0;
        for (int i = 0; i < TILE; ++i) { s_off[i] = o; o += s_len[i]; }
    }

    if (tid < 32) {                  // wave 0 only: tensor ops ignore EXEC,
                                     // so gate by wave to issue exactly one DMA
        // Build the sorted 16-bit row-index list (strictly increasing — the
        // TDM gather-mode OOB requirement).
        unsigned short idx[MAXROWS];
        int n = 0;
        for (int i = 0; i < TILE; ++i)
            for (int j = 0; j < s_len[i]; ++j)
                idx[n++] = (unsigned short)(s_start[i] + j);
        const unsigned short last = n ? idx[n - 1] : 0;
        for (int k = n; k < MAXROWS; ++k) idx[k] = last;  // masked by tile_dim1

        if (n > 0) {
            const uint64_t gaddr =
                (uint64_t)(uintptr_t)enc + (uint64_t)b * Pv * Hv * 4ull;
            const uint32_t laddr = (uint32_t)(uintptr_t)(&rows_lds[0][0]);

            // D# group 0 (ISA 8.3): count=1 | gather_mode(bit31) |
            // 16-bit indices(bit30=0); lds_addr; global_addr[56:0]; type=2.
            u32x4 g0;
            g0.x = 0x80000001u;
            g0.y = laddr;
            g0.z = (uint32_t)gaddr;
            g0.w = ((uint32_t)(gaddr >> 32) & 0x01FFFFFFu) | (2u << 30);

            // D# group 1 (ISA 8.4): data_size=2 (4B) at [17:16];
            // tensor_dim0=768 [79:48]; tensor_dim1=512 [111:80];
            // tile_dim0=768 [127:112]; tile_dim1=n [143:128];
            // tensor_dim0_stride=768 [207:160]; stride1 ignored in gather.
            i32x8 g1;
            g1[0] = 0x00020000;
            g1[1] = (int)((unsigned)(Hv & 0xFFFF) << 16);
            g1[2] = (int)((unsigned)(Pv & 0xFFFF) << 16);
            g1[3] = (int)((unsigned)(Hv & 0xFFFF) << 16);
            g1[4] = n;
            g1[5] = Hv;
            g1[6] = 0;
            g1[7] = 0;

            // D# groups 2-3 (ISA 8.7, 16-bit index mode): row_index_0..15.
            i32x4 g2, g3;
            g2[0] = (int)((unsigned)idx[0]  | ((unsigned)idx[1]  << 16));
            g2[1] = (int)((unsigned)idx[2]  | ((unsigned)idx[3]  << 16));
            g2[2] = (int)((unsigned)idx[4]  | ((unsigned)idx[5]  << 16));
            g2[3] = (int)((unsigned)idx[6]  | ((unsigned)idx[7]  << 16));
            g3[0] = (int)((unsigned)idx[8]  | ((unsigned)idx[9]  << 16));
            g3[1] = (int)((unsigned)idx[10] | ((unsigned)idx[11] << 16));
            g3[2] = (int)((unsigned)idx[12] | ((unsigned)idx[13] << 16));
            g3[3] = (int)((unsigned)idx[14] | ((unsigned)idx[15] << 16));

#if __clang_major__ >= 23
            i32x8 gz = {0, 0, 0, 0, 0, 0, 0, 0};
            __builtin_amdgcn_tensor_load_to_lds(g0, g1, g2, g3, gz, 0);
#else
            __builtin_amdgcn_tensor_load_to_lds(g0, g1, g2, g3, 0);
#endif
        }
        __builtin_amdgcn_s_wait_tensorcnt(0);
    }
    __syncthreads();   // publish DMA-written LDS to all waves

    // Reduce + store: rows are compacted, token i's subwords at s_off[i].
    float4* __restrict__ out4 = (float4*)out;
#pragma unroll 2
    for (int c = tid; c < TILE * H4v; c += BLOCK) {
        const int i   = c / H4v;
        const int col = c - i * H4v;
        const int l   = s_len[i];
        float4 r = make_float4(0.f, 0.f, 0.f, 0.f);
        if (l > 0) {
            const int p = s_off[i];
            float4 a = rows_lds[p][col];
            if (l > 1) {
                const float4 bb = rows_lds[p + 1][col];
                a.x = (a.x + bb.x) * 0.5f;
                a.y = (a.y + bb.y) * 0.5f;
                a.z = (a.z + bb.z) * 0.5f;
                a.w = (a.w + bb.w) * 0.5f;
            }
            r = a;
        }
        out4[((size_t)b * Tv + t0 + i) * H4v + col] = r;
    }
}

// ---------------------------------------------------------------------------
// Fallback gather kernel: per-lane GLOBAL_LOAD_ASYNC_TO_LDS_B128 (ASYNCcnt).
// Compiled as an alternate variant; swap the launch below to use it if the
// TDM descriptor path proves wrong.
// ---------------------------------------------------------------------------
__device__ __forceinline__ void async_copy_b128(uint32_t lds_off, uint64_t gaddr) {
    asm volatile("global_load_async_to_lds_b128 %0, %1, off"
                 :: "v"(lds_off), "v"(gaddr)
                 : "memory");
}

__global__ __launch_bounds__(BLOCK) void bert_gather_async_kernel(
        const float* __restrict__ enc,
        const int*   __restrict__ lens,
        const int*   __restrict__ start,
        float*       __restrict__ out) {
    __shared__ float4 rows_lds[MAXROWS][H4v];
    __shared__ int s_start[TILE];
    __shared__ int s_len[TILE];

    const int b   = blockIdx.y;
    const int t0  = blockIdx.x * TILE;
    const int tid = threadIdx.x;

    if (tid < TILE) {
        s_start[tid] = start[b * Tv + t0 + tid];
        s_len[tid]   = lens[b * Tv + t0 + tid];
    }
    __syncthreads();

    const uint32_t lds_base = (uint32_t)(uintptr_t)(&rows_lds[0][0]);
    const uint64_t enc_base =
        (uint64_t)(uintptr_t)enc + (uint64_t)b * Pv * Hv * 4ull;

#pragma unroll 4
    for (int c = tid; c < MAXROWS * H4v; c += BLOCK) {
        const int row = c / H4v;
        const int col = c - row * H4v;
        const int i = row >> 1;
        const int j = row & 1;
        if (j < s_len[i]) {
            const uint64_t g =
                enc_base + ((uint64_t)(s_start[i] + j) * Hv + (uint64_t)col * 4) * 4ull;
            async_copy_b128(lds_base + 16u * (uint32_t)(row * H4v + col), g);
        }
    }
    asm volatile("s_wait_asynccnt 0x0" ::: "memory");
    __syncthreads();

    float4* __restrict__ out4 = (float4*)out;
#pragma unroll 2
    for (int c = tid; c < TILE * H4v; c += BLOCK) {
        const int i   = c / H4v;
        const int col = c - i * H4v;
        const int l   = s_len[i];
        float4 r = make_float4(0.f, 0.f, 0.f, 0.f);
        if (l > 0) {
            float4 a = rows_lds[2 * i][col];
            if (l > 1) {
                const float4 bb = rows_lds[2 * i + 1][col];
                a.x = (a.x + bb.x) * 0.5f;
                a.y = (a.y + bb.y) * 0.5f;
                a.z = (a.z + bb.z) * 0.5f;
                a.w = (a.w + bb.w) * 0.5f;
            }
            r = a;
        }
        out4[((size_t)b * Tv + t0 + i) * H4v + col] = r;
    }
}

// ---------------------------------------------------------------------------
// Kernel 1: exclusive prefix sum of bert_lens per batch row (start offsets).
// ---------------------------------------------------------------------------
__global__ __launch_bounds__(Tv) void bert_scan_kernel(
        const int* __restrict__ lens, int* __restrict__ start) {
    __shared__ int s[Tv];
    const int b = blockIdx.x;
    const int t = threadIdx.x;
    const int v = lens[b * Tv + t];
    s[t] = v;
    __syncthreads();
#pragma unroll
    for (int off = 1; off < Tv; off <<= 1) {
        int x = (t >= off) ? s[t - off] : 0;
        __syncthreads();
        s[t] += x;
        __syncthreads();
    }
    start[b * Tv + t] = s[t] - v;   // exclusive prefix
}

// ---------------------------------------------------------------------------
// Launch. Inputs (setup_inputs order): enc_out f32, bert_mask i32 (redundant
// with lens), bert_lens i32. Workspace: BZ*T ints (64 KB) for start offsets.
// ---------------------------------------------------------------------------
extern "C" void kernel_launch(void* const* d_in, const int* in_sizes, int n_in,
                              void* d_out, int out_size, void* d_ws, size_t ws_size,
                              hipStream_t stream) {
    const float* enc   = (const float*)d_in[0];
    const int*   lens  = (const int*)d_in[2];
    float*       out   = (float*)d_out;
    int*         start = (int*)d_ws;   // BZ*T ints = 64 KB

    bert_scan_kernel<<<BZv, Tv, 0, stream>>>(lens, start);
    bert_gather_tdm_kernel<<<dim3(Tv / TILE, BZv), BLOCK, 0, stream>>>(
        enc, lens, start, out);
}